// Syntac_GCN_21509196219028
// MI455X (gfx1250) — compile-verified
//
#include <hip/hip_runtime.h>
#include <hip/hip_bf16.h>

// Syntactic-GCN forward for MI455X (gfx1250, wave32, WMMA).
// Pipeline:
//   1) gemm_qwt : A = q*Wa^T, Bm = q*Wb^T, Hj = q*Wd^T   (V_WMMA_F32_16X16X4_F32)
//   2) edge_softmax : t_ij = w2 . relu(A_i + B_j); masked row softmax -> w
//   3) agg_out : out = relu(q + w*Hj)                    (V_WMMA_F32_16X16X4_F32)

typedef __attribute__((ext_vector_type(2))) float v2f;
typedef __attribute__((ext_vector_type(8))) float v8f;

#define BQ 8
#define LQ 256
#define DQ 256
#define BLD (BQ * LQ * DQ)   // 524288

// ---------------------------------------------------------------------------
// Stage 1: Y[m,c] = sum_k Q[m,k] * W[c,k]   (Y = Q W^T), Q:[M x 256], W:[256 x 256]
// One wave computes one 16x16 output tile; K-loop in steps of 4 via f32 WMMA.
// Fragment layouts per CDNA5 ISA 7.12.2:
//   A 16x4 f32 : lanes 0-15 -> M=l, {K0,K1}; lanes 16-31 -> M=l, {K2,K3}
//   B 4x16 f32 : vgpr0 = rows {K0 | K1} across lane halves, vgpr1 = {K2 | K3}
//   C/D 16x16  : vgpr r -> row (r | r+8) across lane halves, N = lane&15
// ---------------------------------------------------------------------------
__global__ __launch_bounds__(256) void gemm_qwt_kernel(
    const float* __restrict__ Q, const float* __restrict__ W,
    float* __restrict__ Y, int M) {
  const int lane = threadIdx.x & 31;
  const int wave = threadIdx.x >> 5;
  const int half = lane >> 4;     // 0: lanes 0-15, 1: lanes 16-31
  const int l    = lane & 15;

  const int ntN  = DQ / 16;                       // 16 tiles along N
  const int tile = blockIdx.x * 8 + wave;         // 8 waves / block
  const int tM = tile / ntN, tN = tile % ntN;
  if (tM * 16 >= M) return;                       // uniform per wave
  const int m0 = tM * 16, c0 = tN * 16;

  const float* qrow = Q + (size_t)(m0 + l) * DQ;  // A-matrix row for this lane
  const float* wrow = W + (size_t)(c0 + l) * DQ;  // W row c0+l supplies B column

  v8f acc = {};
  #pragma unroll 8
  for (int k0 = 0; k0 < DQ; k0 += 4) {
    v2f a, b;
    a.x = qrow[k0 + 2 * half + 0];
    a.y = qrow[k0 + 2 * half + 1];
    b.x = wrow[k0 + half + 0];      // B[K=half][N=l]   = W[c0+l, k0+half]
    b.y = wrow[k0 + half + 2];      // B[K=half+2][N=l] = W[c0+l, k0+half+2]
    acc = __builtin_amdgcn_wmma_f32_16x16x4_f32(
        false, a, false, b, (short)0, acc, false, false);
  }

  #pragma unroll
  for (int r = 0; r < 8; ++r) {
    const int row = m0 + r + half * 8;
    Y[(size_t)row * DQ + (c0 + l)] = acc[r];
  }
}

// ---------------------------------------------------------------------------
// Stage 2: per (b,i) row: t_j = sum_c relu(A[b,i,c] + Bm[b,j,c]) * w2[c]
// then masked softmax over j; w[b,i,j] = mask * softmax(where(mask, t, -100)).
// Block = 256 threads (one per j). A_i and w2 cached in LDS.
// ---------------------------------------------------------------------------
__global__ __launch_bounds__(256) void edge_softmax_kernel(
    const float* __restrict__ Aact, const float* __restrict__ Bact,
    const float* __restrict__ w2, const int* __restrict__ dep,
    float* __restrict__ wout) {
  __shared__ float sA[DQ];
  __shared__ float sW2[DQ];
  __shared__ float red[256];

  const int bi = blockIdx.x;            // 0 .. B*L-1
  const int b  = bi >> 8;
  const int j  = threadIdx.x;

  sA[j]  = Aact[(size_t)bi * DQ + j];
  sW2[j] = w2[j];
  __syncthreads();

  const float* brow = Bact + ((size_t)(b * LQ + j)) * DQ;
  float s = 0.0f;
  #pragma unroll 4
  for (int c = 0; c < DQ; c += 4) {
    const float4 bm = *reinterpret_cast<const float4*>(brow + c);
    s = fmaf(fmaxf(sA[c + 0] + bm.x, 0.0f), sW2[c + 0], s);
    s = fmaf(fmaxf(sA[c + 1] + bm.y, 0.0f), sW2[c + 1], s);
    s = fmaf(fmaxf(sA[c + 2] + bm.z, 0.0f), sW2[c + 2], s);
    s = fmaf(fmaxf(sA[c + 3] + bm.w, 0.0f), sW2[c + 3], s);
  }

  const int   m = dep[(size_t)bi * LQ + j] > 0;
  const float T = m ? s : -100.0f;

  // row max
  red[j] = T; __syncthreads();
  #pragma unroll
  for (int off = 128; off > 0; off >>= 1) {
    if (j < off) red[j] = fmaxf(red[j], red[j + off]);
    __syncthreads();
  }
  const float mx = red[0];
  __syncthreads();

  const float e = expf(T - mx);
  red[j] = e; __syncthreads();
  #pragma unroll
  for (int off = 128; off > 0; off >>= 1) {
    if (j < off) red[j] += red[j + off];
    __syncthreads();
  }
  const float inv = 1.0f / red[0];

  wout[(size_t)bi * LQ + j] = m ? (e * inv) : 0.0f;
}

// ---------------------------------------------------------------------------
// Stage 3: per batch b: agg = w[b] (LxL) * Hj[b] (LxD); out = relu(q + agg).
// One wave per 16x16 output tile, f32 WMMA, fused relu+residual epilogue.
// ---------------------------------------------------------------------------
__global__ __launch_bounds__(256) void agg_out_kernel(
    const float* __restrict__ Wgt, const float* __restrict__ Hj,
    const float* __restrict__ Q, float* __restrict__ Out) {
  const int lane = threadIdx.x & 31;
  const int wave = threadIdx.x >> 5;
  const int half = lane >> 4;
  const int l    = lane & 15;

  const int tile = blockIdx.x * 8 + wave;   // 2048 tiles total
  const int b    = tile >> 8;               // 256 tiles per batch
  const int rem  = tile & 255;
  const int i0 = (rem >> 4) * 16;           // tile row (i)
  const int c0 = (rem & 15) * 16;           // tile col (c)

  const float* arow = Wgt + ((size_t)b * LQ + (i0 + l)) * LQ;  // w row, K contiguous
  const float* hb   = Hj  + (size_t)b * LQ * DQ;

  v8f acc = {};
  #pragma unroll 8
  for (int k0 = 0; k0 < LQ; k0 += 4) {
    v2f a, bf;
    a.x  = arow[k0 + 2 * half + 0];
    a.y  = arow[k0 + 2 * half + 1];
    bf.x = hb[(size_t)(k0 + half + 0) * DQ + (c0 + l)];   // B[K][N] = Hj[j=K, c=N]
    bf.y = hb[(size_t)(k0 + half + 2) * DQ + (c0 + l)];
    acc = __builtin_amdgcn_wmma_f32_16x16x4_f32(
        false, a, false, bf, (short)0, acc, false, false);
  }

  #pragma unroll
  for (int r = 0; r < 8; ++r) {
    const int row = i0 + r + half * 8;
    const size_t idx = (size_t)b * LQ * DQ + (size_t)row * DQ + (c0 + l);
    Out[idx] = fmaxf(Q[idx] + acc[r], 0.0f);
  }
}

// ---------------------------------------------------------------------------
extern "C" void kernel_launch(void* const* d_in, const int* in_sizes, int n_in,
                              void* d_out, int out_size, void* d_ws, size_t ws_size,
                              hipStream_t stream) {
  const float* q        = (const float*)d_in[0];   // [B,L,D] f32
  const int*   wordlens = (const int*)d_in[1];     // [B] i32
  const int*   dep      = (const int*)d_in[2];     // [B,L,L] i32
  const float* Wa       = (const float*)d_in[3];   // [D,D]
  const float* Wb       = (const float*)d_in[4];   // [D,D]
  const float* w2       = (const float*)d_in[5];   // [D]
  const float* Wd       = (const float*)d_in[6];   // [D,D]
  float* out = (float*)d_out;

  float* Aact = (float*)d_ws;          // B*L*D
  float* Bact = Aact + BLD;            // B*L*D
  float* Hjw  = Bact + BLD;            // B*L*D
  float* wgt  = Hjw + BLD;             // B*L*L

  const dim3 blk(256);
  // 2048 rows x 256 cols -> 2048 tiles -> 256 blocks of 8 waves
  gemm_qwt_kernel<<<256, blk, 0, stream>>>(q, Wa, Aact, BQ * LQ);
  gemm_qwt_kernel<<<256, blk, 0, stream>>>(q, Wb, Bact, BQ * LQ);
  gemm_qwt_kernel<<<256, blk, 0, stream>>>(q, Wd, Hjw, BQ * LQ);

  edge_softmax_kernel<<<BQ * LQ, blk, 0, stream>>>(Aact, Bact, w2, dep, wgt);

  agg_out_kernel<<<256, blk, 0, stream>>>(wgt, Hjw, q, out);

  // Tuple outputs 2 & 3: raw pass-through (bit-copy ints into the flat buffer).
  hipMemcpyAsync(out + BLD, wordlens, (size_t)BQ * sizeof(int),
                 hipMemcpyDeviceToDevice, stream);
  hipMemcpyAsync(out + BLD + BQ, dep, (size_t)BQ * LQ * LQ * sizeof(int),
                 hipMemcpyDeviceToDevice, stream);
}